// GELU231_23648089932113
// MI455X (gfx1250) — compile-verified
//
#include <hip/hip_runtime.h>
#include <math.h>

// Problem constants (from reference): B=4, T=4096, D=1024, N=512
#define DIMS   1024
#define NROWS  16384   // B*T
#define NBUF   512

typedef float v2f __attribute__((ext_vector_type(2)));
typedef float v8f __attribute__((ext_vector_type(8)));

// Branch-free tanh: prefer the gfx1250 hardware transcendental V_TANH_F32.
__device__ __forceinline__ float fast_tanh(float u) {
#if __has_builtin(__builtin_amdgcn_tanhf)
    return __builtin_amdgcn_tanhf(u);
#else
    float a = fabsf(u);
    float e = __expf(-2.0f * a);            // underflows to 0 for large a -> th=1
    float th = (1.0f - e) / (1.0f + e);
    return copysignf(th, u);
#endif
}

__device__ __forceinline__ float gelu1(float x) {
    const float C = 0.7978845608028654f; // sqrt(2/pi)
    float u = C * (x + 0.044715f * x * x * x);
    return 0.5f * x * (1.0f + fast_tanh(u));
}
__device__ __forceinline__ float4 gelu4(float4 v) {
    float4 r;
    r.x = gelu1(v.x); r.y = gelu1(v.y); r.z = gelu1(v.z); r.w = gelu1(v.w);
    return r;
}

// ---------------------------------------------------------------------------
// Kernel 1: per-block partial column sums of gelu(x).
// grid=256 blocks, block=256 threads. Block b handles rows [b*64, b*64+64).
// Thread t owns columns [4t, 4t+4) -> fully coalesced float4 row reads.
// Deterministic (fixed accumulation order per thread).
// ---------------------------------------------------------------------------
__global__ void k_colsum(const float* __restrict__ x, float* __restrict__ partial) {
    int t = threadIdx.x;
    int b = blockIdx.x;
    const float4* x4 = (const float4*)x;
    float4 acc = make_float4(0.f, 0.f, 0.f, 0.f);
    int r0 = b * 64;
    for (int r = 0; r < 64; ++r) {
        float4 g = gelu4(x4[(size_t)(r0 + r) * 256 + t]);
        acc.x += g.x; acc.y += g.y; acc.z += g.z; acc.w += g.w;
    }
    ((float4*)partial)[b * 256 + t] = acc;
}

// ---------------------------------------------------------------------------
// Kernel 1b: reduce 256 partials -> m_curr (column mean). grid=4, block=256.
// ---------------------------------------------------------------------------
__global__ void k_colreduce(const float* __restrict__ partial, float* __restrict__ m_curr) {
    int d = blockIdx.x * 256 + threadIdx.x;
    float s = 0.f;
    for (int b = 0; b < 256; ++b) s += partial[b * DIMS + d];
    m_curr[d] = s * (1.0f / (float)NROWS);
}

// ---------------------------------------------------------------------------
// Copy buf -> new_buf region of d_out (row `ptr` overwritten later by k_recall).
// grid=512, block=256, float4.
// ---------------------------------------------------------------------------
__global__ void k_copybuf(const float4* __restrict__ buf, float4* __restrict__ nbuf) {
    int i = blockIdx.x * 256 + threadIdx.x;
    nbuf[i] = buf[i];
}

// ---------------------------------------------------------------------------
// Kernel 2 (single block, 256 threads = 8 waves): recall path.
//  - m_norm = normalize(m_curr); write into new_buf row `ptr`
//  - sims over 512 buf rows (wave-parallel dot + shfl reduction), masked argmax
//  - nv_n = normalize(buf[argmax]) -> ws
//  - tau = exp(log_tau), alpha = sigmoid(log_blend) -> ws scalars
// All reductions fixed-order -> deterministic.
// ---------------------------------------------------------------------------
__global__ void k_recall(const float* __restrict__ m_curr,
                         const float* __restrict__ buf,
                         const unsigned char* __restrict__ mask,
                         const int* __restrict__ ptr,
                         const float* __restrict__ log_tau,
                         const float* __restrict__ log_blend,
                         float* __restrict__ new_buf,
                         float* __restrict__ nv_n,
                         float* __restrict__ scal) {
    __shared__ float s_m[DIMS];
    __shared__ float s_red[256];
    __shared__ float s_best[8];
    __shared__ int   s_bidx[8];
    __shared__ int   s_near;

    int t = threadIdx.x;

    // ||m_curr||
    float p = 0.f;
    for (int i = 0; i < 4; ++i) { float v = m_curr[t + 256 * i]; p += v * v; }
    s_red[t] = p; __syncthreads();
    for (int s = 128; s > 0; s >>= 1) { if (t < s) s_red[t] += s_red[t + s]; __syncthreads(); }
    float inv = 1.0f / fmaxf(sqrtf(s_red[0]), 1e-12f);

    int pv = ptr[0];
    for (int i = 0; i < 4; ++i) {
        int d = t + 256 * i;
        float mn = m_curr[d] * inv;
        s_m[d] = mn;
        new_buf[(size_t)pv * DIMS + d] = mn;   // ring-buffer scatter write
    }
    __syncthreads();

    // sims + argmax: wave w handles rows [w*64, w*64+64) in order
    int w = t >> 5, l = t & 31;
    float best = -2.0f; int bidx = 0;
    for (int i = 0; i < 64; ++i) {
        int n = w * 64 + i;
        const float* row = buf + (size_t)n * DIMS;
        float dot = 0.f, n2 = 0.f;
        for (int j = 0; j < 32; ++j) {
            float v = row[l + 32 * j];
            dot += v * s_m[l + 32 * j];
            n2  += v * v;
        }
        for (int off = 16; off > 0; off >>= 1) {
            dot += __shfl_xor(dot, off, 32);
            n2  += __shfl_xor(n2,  off, 32);
        }
        float sim = dot / fmaxf(sqrtf(n2), 1e-12f);
        sim = mask[n] ? sim : -1.0f;
        if (sim > best) { best = sim; bidx = n; }   // strict > keeps earliest index
    }
    if (l == 0) { s_best[w] = best; s_bidx[w] = bidx; }
    __syncthreads();
    if (t == 0) {
        float bb = s_best[0]; int bi = s_bidx[0];
        for (int wv = 1; wv < 8; ++wv)
            if (s_best[wv] > bb) { bb = s_best[wv]; bi = s_bidx[wv]; }
        s_near = bi;
        scal[0] = expf(log_tau[0]);                      // tau
        scal[1] = 1.0f / (1.0f + expf(-log_blend[0]));   // alpha
    }
    __syncthreads();

    // nv_n = normalize(buf[nearest])
    const float* nrow = buf + (size_t)s_near * DIMS;
    p = 0.f;
    for (int i = 0; i < 4; ++i) { float v = nrow[t + 256 * i]; p += v * v; }
    s_red[t] = p; __syncthreads();
    for (int s = 128; s > 0; s >>= 1) { if (t < s) s_red[t] += s_red[t + s]; __syncthreads(); }
    float inv2 = 1.0f / fmaxf(sqrtf(s_red[0]), 1e-12f);
    for (int i = 0; i < 4; ++i) { int d = t + 256 * i; nv_n[d] = nrow[d] * inv2; }
}

// ---------------------------------------------------------------------------
// Kernel 3 (main fused pass): 1024 blocks x 256 threads; block = 16-token tile.
// Per wave w (8 waves): K-slice [w*128, w*128+128) in 32 steps of K=4.
//   WMMA chain 1 (sim):  A = gelu(x) fragment, B = nv_n broadcast to 16 cols
//   WMMA chain 2 (gram): A = B = gelu(x) fragment  (A/B lane layouts coincide:
//       A(16x4): m=lane%16, k=vgpr+2*(lane/16); B(4x16): n=lane%16, same k map)
//   -> D_sim[m][*] = out[m].nv ; diag(D_gram)[m] = ||out[m]||^2   (full fp32)
// Partial 16x16 accumulators reduced across 8 waves in LDS, gates computed,
// then coalesced gelu+scale+store (x re-read from the 192MB L2).
// Control flow is uniform -> EXEC all-ones at every WMMA (ISA requirement).
// ---------------------------------------------------------------------------
__global__ void k_gate(const float* __restrict__ x,
                       const float* __restrict__ nv_n,
                       const float* __restrict__ scal,
                       float* __restrict__ out) {
    __shared__ float s_nv[DIMS];
    __shared__ float s_sim[8 * 256];
    __shared__ float s_gram[8 * 256];
    __shared__ float s_gate[16];

    int t = threadIdx.x;
    for (int i = 0; i < 4; ++i) s_nv[t + 256 * i] = nv_n[t + 256 * i];
    __syncthreads();

    int w = t >> 5, l = t & 31;
    int row  = l & 15;          // token within tile
    int half = l >> 4;          // lane half selects k-pair {0,1} vs {2,3}
    size_t tile0 = (size_t)blockIdx.x * 16;
    const float* xrow = x + (tile0 + row) * DIMS;

    v8f simc  = {0.f, 0.f, 0.f, 0.f, 0.f, 0.f, 0.f, 0.f};
    v8f gramc = {0.f, 0.f, 0.f, 0.f, 0.f, 0.f, 0.f, 0.f};

    int kb = w * 128 + 2 * half;
    const v2f* __restrict__ xp = reinterpret_cast<const v2f*>(xrow + kb);
    const v2f* __restrict__ bp = reinterpret_cast<const v2f*>(s_nv + kb);

#pragma unroll 8
    for (int s = 0; s < 32; ++s) {
        v2f raw = xp[2 * s];            // global_load_b64, immediate offset
        v2f bv  = bp[2 * s];            // ds_load_b64, LDS broadcast per half
        v2f av; av[0] = gelu1(raw[0]); av[1] = gelu1(raw[1]);
        simc = __builtin_amdgcn_wmma_f32_16x16x4_f32(
            false, av, false, bv, (short)0, simc, false, false);
        gramc = __builtin_amdgcn_wmma_f32_16x16x4_f32(
            false, av, false, av, (short)0, gramc, false, false);
    }

    // dump per-wave partial accumulators: slot = vgpr*32 + lane
    for (int v = 0; v < 8; ++v) {
        s_sim [w * 256 + v * 32 + l] = simc[v];
        s_gram[w * 256 + v * 32 + l] = gramc[v];
    }
    __syncthreads();

    // cross-wave sum (thread t exclusively owns slot t)
    float st = 0.f, gt = 0.f;
    for (int wv = 0; wv < 8; ++wv) {
        st += s_sim [wv * 256 + t];
        gt += s_gram[wv * 256 + t];
    }
    s_sim[t] = st; s_gram[t] = gt;
    __syncthreads();

    // gates for the 16 tokens
    if (t < 16) {
        int m = t;
        int vs = m & 7;
        int lane_sim  = (m < 8) ? 0 : 16;        // any column N (all equal)
        int lane_gram = (m < 8) ? m : (m + 16);  // diagonal N==m
        float sim = s_sim [vs * 32 + lane_sim];
        float g2  = s_gram[vs * 32 + lane_gram];
        float nrm = sqrtf(fmaxf(g2, 0.0f));
        float tok = sim / fmaxf(nrm, 1e-12f);
        float tau = scal[0], alpha = scal[1];
        s_gate[m] = 1.0f - alpha + alpha * expf(-tau * tok);
    }
    __syncthreads();

    // coalesced gelu + gate + store (x served from L2 on this second touch)
    const float4* x4 = (const float4*)(x + tile0 * DIMS);
    float4*       o4 = (float4*)(out + tile0 * DIMS);
    for (int j = 0; j < 16; ++j) {
        float4 g = gelu4(x4[j * 256 + t]);
        float gg = s_gate[j];
        g.x *= gg; g.y *= gg; g.z *= gg; g.w *= gg;
        o4[j * 256 + t] = g;
    }
}

// ---------------------------------------------------------------------------
extern "C" void kernel_launch(void* const* d_in, const int* in_sizes, int n_in,
                              void* d_out, int out_size, void* d_ws, size_t ws_size,
                              hipStream_t stream) {
    (void)in_sizes; (void)n_in; (void)out_size; (void)ws_size;

    const float*         x         = (const float*)d_in[0];         // [4,4096,1024]
    const float*         buf       = (const float*)d_in[1];         // [512,1024]
    const unsigned char* mask      = (const unsigned char*)d_in[2]; // [512] bool
    const int*           ptr       = (const int*)d_in[3];           // scalar
    const float*         log_tau   = (const float*)d_in[4];         // scalar
    const float*         log_blend = (const float*)d_in[5];         // scalar

    float* out     = (float*)d_out;                      // [16777216]
    float* new_buf = out + (size_t)NROWS * DIMS;         // [524288]

    float* ws      = (float*)d_ws;
    float* partial = ws;                 // 256*1024 floats
    float* m_curr  = ws + 256 * DIMS;    // 1024
    float* nv      = m_curr + DIMS;      // 1024
    float* scal    = nv + DIMS;          // 2 (tau, alpha)

    k_colsum   <<<256,  256, 0, stream>>>(x, partial);
    k_colreduce<<<4,    256, 0, stream>>>(partial, m_curr);
    k_copybuf  <<<512,  256, 0, stream>>>((const float4*)buf, (float4*)new_buf);
    k_recall   <<<1,    256, 0, stream>>>(m_curr, buf, mask, ptr, log_tau, log_blend,
                                          new_buf, nv, scal);
    k_gate     <<<1024, 256, 0, stream>>>(x, nv, scal, out);
}